// CausalODE_39719857553826
// MI455X (gfx1250) — compile-verified
//
#include <hip/hip_runtime.h>

// ---------------------------------------------------------------------------
// CausalODE for MI455X (gfx1250, wave32, WMMA).
// Phase A: serial RK4 scan (1 WGP-resident block).  Phase B: decoder GEMMs
//          writing West/Mlag in a pair-swizzled layout.
// Phase C: batched (x @ West[t]) + (x_prev @ Mlag) via V_WMMA_F32_16X16X4_F32;
//          tiles staged to LDS with global_load_async_to_lds_b128; B fragments
//          are single ds_load_b64s thanks to the swizzle; NT stores for out.
// ---------------------------------------------------------------------------

#define USE_ASYNC_LDS 1

typedef float v2f __attribute__((ext_vector_type(2)));
typedef float v8f __attribute__((ext_vector_type(8)));

constexpr int Dd  = 64;
constexpr int Kk  = 32;
constexpr int Bb  = 4096;
constexpr int Tt  = 64;
constexpr int DK  = 2048;   // D*K
constexpr int HID = 4096;   // 2*DK
constexpr int Hh  = 512;

// pair-swizzled 64x64 tile: elem (k, n) -> (k>>1)*128 + n*2 + (k&1)
__host__ __device__ __forceinline__ int swz(int k, int n) {
  return ((k >> 1) << 7) + (n << 1) + (k & 1);
}

// ---- workspace layout (float offsets) ----
constexpr size_t OFF_ZTRAJ = 0;                                // T*DK
constexpr size_t OFF_TH    = OFF_ZTRAJ + (size_t)Tt * DK;      // T*DK
constexpr size_t OFF_H1    = OFF_TH    + (size_t)Tt * DK;      // T*H
constexpr size_t OFF_S2    = OFF_H1    + (size_t)Tt * Hh;      // T*H
constexpr size_t OFF_WEST  = OFF_S2    + (size_t)Tt * Hh;      // T*D*D (swizzled)
constexpr size_t OFF_MLAG  = OFF_WEST  + (size_t)Tt * Dd * Dd; // D*D   (swizzled)
constexpr size_t OFF_SCR   = OFF_MLAG  + (size_t)Dd * Dd;
constexpr size_t SC_HID  = OFF_SCR;            // 4096
constexpr size_t SC_XLIN = SC_HID  + HID;      // 2048
constexpr size_t SC_THZ  = SC_XLIN + DK;       // 2048
constexpr size_t SC_G    = SC_THZ  + DK;       // 2048
constexpr size_t SC_W    = SC_G    + DK;       // 4096
constexpr size_t SC_M    = SC_W    + 4096;     // 4096
constexpr size_t SC_EA   = SC_M    + 4096;     // 4096
constexpr size_t SC_EB   = SC_EA   + 4096;     // 4096
constexpr size_t SC_TA   = SC_EB   + 4096;     // 4096
constexpr size_t SC_TB   = SC_TA   + 4096;     // 4096
// total ws need ~= (SC_TB + 4096) * 4 bytes  (~2.6 MB)

// ---------------------------------------------------------------------------
// Phase A: the ODE right-hand side evaluated block-cooperatively.
// ---------------------------------------------------------------------------
struct AP {
  const float *l1w, *l1b, *l2w, *l2b;
  const float *d1w, *d1b, *d2w, *d2b, *d3w, *d3b;
  float *hid, *xlin, *thz, *g;          // global scratch
  float *wm, *Mm, *Ea, *Eb, *Ta, *Tb;   // 64x64 matrices (global scratch)
  float *sa, *sb, *sc, *red, *scal;     // shared scratch
};

__device__ void ode_func(const AP& p, float t, const float* zin, float* out) {
  const int tid = threadIdx.x;
  const int NT  = blockDim.x;

  // hid = tanh(z @ l1_w^T + l1_b)
  for (int j = tid; j < HID; j += NT) {
    const float* wr = p.l1w + (size_t)j * DK;
    float acc = p.l1b[j];
    for (int i = 0; i < DK; ++i) acc = fmaf(zin[i], wr[i], acc);
    p.hid[j] = tanhf(acc);
  }
  __syncthreads();
  // xlin = hid @ l2_w^T + l2_b ;  thz = tanh(xlin)
  for (int j = tid; j < DK; j += NT) {
    const float* wr = p.l2w + (size_t)j * HID;
    float acc = p.l2b[j];
    for (int i = 0; i < HID; ++i) acc = fmaf(p.hid[i], wr[i], acc);
    p.xlin[j] = acc;
    p.thz[j]  = tanhf(acc);
  }
  __syncthreads();
  // decoder layer 1 (input = [thz, t])
  for (int a = tid; a < Hh; a += NT) {
    const float* wr = p.d1w + (size_t)a * (DK + 1);
    float acc = fmaf(t, wr[DK], p.d1b[a]);
    for (int i = 0; i < DK; ++i) acc = fmaf(p.thz[i], wr[i], acc);
    p.sa[a] = acc;
  }
  __syncthreads();
  // decoder layer 2 + silu
  for (int c = tid; c < Hh; c += NT) {
    const float* wr = p.d2w + (size_t)c * Hh;
    float acc = p.d2b[c];
    for (int i = 0; i < Hh; ++i) acc = fmaf(p.sa[i], wr[i], acc);
    p.sb[c] = acc;                               // pre-activation (for silu')
    float s = 1.f / (1.f + expf(-acc));
    p.sc[c] = acc * s;                           // silu
  }
  __syncthreads();
  // decoder layer 3 -> w (64x64),  M = w .* w
  for (int q = tid; q < Dd * Dd; q += NT) {
    const float* wr = p.d3w + (size_t)q * Hh;
    float acc = p.d3b[q];
    for (int i = 0; i < Hh; ++i) acc = fmaf(p.sc[i], wr[i], acc);
    p.wm[q] = acc;
    p.Mm[q] = acc * acc;
  }
  __syncthreads();

  // ---- expm(M): scaling & squaring + order-12 Taylor ----
  if (tid < Dd) {
    float rs = 0.f;
    for (int c = 0; c < Dd; ++c) rs += fabsf(p.Mm[tid * Dd + c]);
    p.red[tid] = rs;
  }
  __syncthreads();
  if (tid == 0) {
    float mx = 0.f;
    for (int i = 0; i < Dd; ++i) mx = fmaxf(mx, p.red[i]);
    int s = 0;
    if (mx > 0.25f) { s = (int)ceilf(log2f(mx)) + 1; if (s < 0) s = 0; if (s > 30) s = 30; }
    p.scal[0] = (float)s;
  }
  __syncthreads();
  const int   sq  = (int)p.scal[0];
  const float scl = ldexpf(1.f, -sq);
  for (int q = tid; q < Dd * Dd; q += NT) {
    float a = p.Mm[q] * scl;
    p.Mm[q] = a;                                          // A
    p.Ta[q] = a;                                          // term = A
    p.Ea[q] = a + (((q >> 6) == (q & 63)) ? 1.f : 0.f);   // E = I + A
  }
  __syncthreads();
  float* term = p.Ta; float* tn = p.Tb;
  for (int k = 2; k <= 12; ++k) {
    const float invk = 1.f / (float)k;
    for (int q = tid; q < Dd * Dd; q += NT) {
      int r = q >> 6, c = q & 63;
      const float* tr_ = term + r * Dd;
      float acc = 0.f;
      for (int u = 0; u < Dd; ++u) acc = fmaf(tr_[u], p.Mm[u * Dd + c], acc);
      tn[q] = acc * invk;
    }
    __syncthreads();
    for (int q = tid; q < Dd * Dd; q += NT) p.Ea[q] += tn[q];
    __syncthreads();
    float* tmp = term; term = tn; tn = tmp;
  }
  float* E = p.Ea; float* E2 = p.Eb;
  for (int it = 0; it < sq; ++it) {
    for (int q = tid; q < Dd * Dd; q += NT) {
      int r = q >> 6, c = q & 63;
      const float* er = E + r * Dd;
      float acc = 0.f;
      for (int u = 0; u < Dd; ++u) acc = fmaf(er[u], E[u * Dd + c], acc);
      E2[q] = acc;
    }
    __syncthreads();
    float* tmp = E; E = E2; E2 = tmp;
  }
  if (tid == 0) {
    float tr = 0.f;
    for (int i = 0; i < Dd; ++i) tr += E[i * Dd + i];
    p.scal[1] = tr - (float)Dd;                   // h = trace(expm(M)) - d
  }
  __syncthreads();

  // ---- gradient: d trace(expm(M))/dM = expm(M)^T ; chain back to z ----
  float* gw = (E == p.Ea) ? p.Eb : p.Ea;
  for (int q = tid; q < Dd * Dd; q += NT) {
    int r = q >> 6, c = q & 63;
    gw[q] = 2.f * p.wm[q] * E[c * Dd + r];
  }
  __syncthreads();
  // gs = gw @ dec3_w
  for (int c = tid; c < Hh; c += NT) {
    float acc = 0.f;
    for (int q = 0; q < Dd * Dd; ++q) acc = fmaf(gw[q], p.d3w[(size_t)q * Hh + c], acc);
    p.sc[c] = acc;
  }
  __syncthreads();
  // gh2 = gs * silu'(h2)
  for (int c = tid; c < Hh; c += NT) {
    float xx = p.sb[c];
    float s = 1.f / (1.f + expf(-xx));
    p.sb[c] = p.sc[c] * (s * (1.f + xx * (1.f - s)));
  }
  __syncthreads();
  // gh1 = gh2 @ dec2_w
  for (int a = tid; a < Hh; a += NT) {
    float acc = 0.f;
    for (int c = 0; c < Hh; ++c) acc = fmaf(p.sb[c], p.d2w[(size_t)c * Hh + a], acc);
    p.sa[a] = acc;
  }
  __syncthreads();
  // g = (gh1 @ dec1_w)[:DK] * (1 - thz^2)
  for (int i = tid; i < DK; i += NT) {
    float acc = 0.f;
    for (int a = 0; a < Hh; ++a) acc = fmaf(p.sa[a], p.d1w[(size_t)a * (DK + 1) + i], acc);
    float th = p.thz[i];
    p.g[i] = acc * (1.f - th * th);
  }
  __syncthreads();
  // gg = sum g^2  (deterministic shuffle-tree + serial combine)
  float loc = 0.f;
  for (int i = tid; i < DK; i += NT) { float gv = p.g[i]; loc = fmaf(gv, gv, loc); }
  for (int off = 16; off > 0; off >>= 1) loc += __shfl_down(loc, off);
  const int lane = tid & 31, wv = tid >> 5;
  if (lane == 0) p.red[wv] = loc;
  __syncthreads();
  if (tid == 0) {
    float gg = 0.f; int nw = NT >> 5;
    for (int i = 0; i < nw; ++i) gg += p.red[i];
    p.scal[2] = gg;
  }
  __syncthreads();
  const float gg = p.scal[2], hval = p.scal[1];
  const float inv  = (gg > 1e-30f) ? 1.f / fmaxf(gg, 1e-30f) : 0.f;
  const float coef = inv * hval;
  for (int i = tid; i < DK; i += NT) out[i] = p.xlin[i] - p.g[i] * coef;
  __syncthreads();
}

__global__ __launch_bounds__(1024)
void k_odeint(const float* iit, const float* iis, const float* encw, const float* encb,
              const float* l1w, const float* l1b, const float* l2w, const float* l2b,
              const float* d1w, const float* d1b, const float* d2w, const float* d2b,
              const float* d3w, const float* d3b, float* ws) {
  __shared__ float sh_z[DK];
  __shared__ float sh_zt[DK];
  __shared__ float sh_k[4][DK];
  __shared__ float sh_a[Hh], sh_b[Hh], sh_c[Hh];
  __shared__ float red[64];
  __shared__ float scal[4];
  const int tid = threadIdx.x, NT = blockDim.x;

  AP p;
  p.l1w = l1w; p.l1b = l1b; p.l2w = l2w; p.l2b = l2b;
  p.d1w = d1w; p.d1b = d1b; p.d2w = d2w; p.d2b = d2b; p.d3w = d3w; p.d3b = d3b;
  p.hid = ws + SC_HID; p.xlin = ws + SC_XLIN; p.thz = ws + SC_THZ; p.g = ws + SC_G;
  p.wm = ws + SC_W; p.Mm = ws + SC_M; p.Ea = ws + SC_EA; p.Eb = ws + SC_EB;
  p.Ta = ws + SC_TA; p.Tb = ws + SC_TB;
  p.sa = sh_a; p.sb = sh_b; p.sc = sh_c; p.red = red; p.scal = scal;

  // init_intra = init_intra_t @ init_intra_s  (into wm)
  for (int q = tid; q < Dd * Dd; q += NT) {
    int r = q >> 6, c = q & 63;
    float acc = 0.f;
    for (int u = 0; u < Kk; ++u) acc = fmaf(iit[r * Kk + u], iis[u * Dd + c], acc);
    p.wm[q] = acc;
  }
  __syncthreads();
  // z0 = relu([init_intra, init_intra^T] @ enc_w^T + enc_b)
  for (int q = tid; q < DK; q += NT) {
    int r = q >> 5, k = q & 31;
    const float* er = encw + (size_t)k * (2 * Dd);
    float acc = encb[k];
    for (int c = 0; c < Dd; ++c) acc = fmaf(p.wm[r * Dd + c], er[c], acc);
    for (int c = 0; c < Dd; ++c) acc = fmaf(p.wm[c * Dd + r], er[Dd + c], acc);
    sh_z[q] = fmaxf(acc, 0.f);
  }
  __syncthreads();
  for (int q = tid; q < DK; q += NT) ws[OFF_ZTRAJ + q] = sh_z[q];
  __syncthreads();

  const float third = 1.f / 3.f;
  for (int step = 0; step < Tt - 1; ++step) {   // RK4 (3/8 rule), dt = 1
    float t0 = (float)(step + 1);
    ode_func(p, t0, sh_z, sh_k[0]);
    for (int q = tid; q < DK; q += NT) sh_zt[q] = fmaf(sh_k[0][q], third, sh_z[q]);
    __syncthreads();
    ode_func(p, t0 + third, sh_zt, sh_k[1]);
    for (int q = tid; q < DK; q += NT) sh_zt[q] = sh_z[q] + (sh_k[1][q] - sh_k[0][q] * third);
    __syncthreads();
    ode_func(p, t0 + 2.f * third, sh_zt, sh_k[2]);
    for (int q = tid; q < DK; q += NT) sh_zt[q] = sh_z[q] + (sh_k[0][q] - sh_k[1][q] + sh_k[2][q]);
    __syncthreads();
    ode_func(p, t0 + 1.f, sh_zt, sh_k[3]);
    for (int q = tid; q < DK; q += NT)
      sh_z[q] = sh_z[q] + (sh_k[0][q] + 3.f * (sh_k[1][q] + sh_k[2][q]) + sh_k[3][q]) * 0.125f;
    __syncthreads();
    for (int q = tid; q < DK; q += NT) ws[OFF_ZTRAJ + (size_t)(step + 1) * DK + q] = sh_z[q];
    __syncthreads();
  }
}

// ---------------------------------------------------------------------------
// Phase B: decoder over the whole trajectory -> West[t] (pair-swizzled), Mlag.
// ---------------------------------------------------------------------------
__global__ void k_tanh(float* ws) {
  int idx = blockIdx.x * blockDim.x + threadIdx.x;
  if (idx < Tt * DK) ws[OFF_TH + idx] = tanhf(ws[OFF_ZTRAJ + idx]);
}

__global__ void k_dec1(const float* d1w, const float* d1b, float* ws) {
  int idx = blockIdx.x * blockDim.x + threadIdx.x;
  if (idx >= Tt * Hh) return;
  int t = idx >> 9, a = idx & (Hh - 1);
  const float* wr = d1w + (size_t)a * (DK + 1);
  const float* th = ws + OFF_TH + (size_t)t * DK;
  float acc = fmaf((float)(t + 1), wr[DK], d1b[a]);   // tgrid = linspace(1, T)
  for (int i = 0; i < DK; ++i) acc = fmaf(th[i], wr[i], acc);
  ws[OFF_H1 + idx] = acc;
}

__global__ void k_dec2(const float* d2w, const float* d2b, float* ws) {
  int idx = blockIdx.x * blockDim.x + threadIdx.x;
  if (idx >= Tt * Hh) return;
  int t = idx >> 9, c = idx & (Hh - 1);
  const float* wr = d2w + (size_t)c * Hh;
  const float* h1 = ws + OFF_H1 + (size_t)t * Hh;
  float acc = d2b[c];
  for (int i = 0; i < Hh; ++i) acc = fmaf(h1[i], wr[i], acc);
  float s = 1.f / (1.f + expf(-acc));
  ws[OFF_S2 + idx] = acc * s;                         // silu
}

__global__ void k_dec3(const float* d3w, const float* d3b, float* ws) {
  int idx = blockIdx.x * blockDim.x + threadIdx.x;
  if (idx >= Tt * Dd * Dd) return;
  int t = idx >> 12, q = idx & (Dd * Dd - 1);
  const float* wr = d3w + (size_t)q * Hh;
  const float* s2 = ws + OFF_S2 + (size_t)t * Hh;
  float acc = d3b[q];
  for (int i = 0; i < Hh; ++i) acc = fmaf(s2[i], wr[i], acc);
  ws[OFF_WEST + (size_t)t * Dd * Dd + swz(q >> 6, q & 63)] = acc;  // swizzled
}

__global__ void k_mlag(const float* uw, const float* vw, float* ws) {
  int q = blockIdx.x * blockDim.x + threadIdx.x;
  if (q >= Dd * Dd) return;
  int e = q >> 6, d = q & 63;
  float acc = 0.f;
  for (int k = 0; k < Kk; ++k) acc = fmaf(uw[k * Dd + e], vw[d * Kk + k], acc);
  ws[OFF_MLAG + swz(e, d)] = acc;                     // Mlag = u_w^T @ v_w^T
}

// ---------------------------------------------------------------------------
// Phase C: out[b,t,:] = x[b,t,:] @ West[t] + x[b,t-1,:] @ Mlag   (fp32 WMMA)
// Grid: (Tcount, B/128); 256 threads = 8 waves; one 16x64 output tile / wave.
// B fragments: one ds_load_b64 each (pair-swizzled LDS tiles); NT stores out.
// ---------------------------------------------------------------------------
template <bool HAS_LAG>
__global__ __launch_bounds__(256)
void k_apply(const float* __restrict__ x, const float* __restrict__ ws,
             float* __restrict__ out, int tbase) {
  const int t     = tbase + blockIdx.x;
  const int brow0 = blockIdx.y * 128;

  __shared__ __align__(16) float sw[Dd * Dd];   // West[t] (swizzled)
  __shared__ __align__(16) float sl[Dd * Dd];   // Mlag    (swizzled)
  const float* west = ws + OFF_WEST + (size_t)t * Dd * Dd;
  const float* mlag = ws + OFF_MLAG;

#if USE_ASYNC_LDS
  {
    // Per-lane async copy: each lane moves 16B/pass, 4 passes = 4096 floats.
    const int base = threadIdx.x * 4;
#pragma unroll
    for (int pass = 0; pass < 4; ++pass) {
      const int fo = base + pass * 1024;
      unsigned ldsw = (unsigned)(uintptr_t)(&sw[fo]);
      const float* gp = west + fo;
      asm volatile("global_load_async_to_lds_b128 %0, %1, off"
                   :: "v"(ldsw), "v"(gp) : "memory");
      if (HAS_LAG) {
        unsigned ldsl = (unsigned)(uintptr_t)(&sl[fo]);
        const float* gq = mlag + fo;
        asm volatile("global_load_async_to_lds_b128 %0, %1, off"
                     :: "v"(ldsl), "v"(gq) : "memory");
      }
    }
    asm volatile("s_wait_asynccnt 0" ::: "memory");
  }
#else
  for (int q = threadIdx.x; q < Dd * Dd; q += 256) {
    sw[q] = west[q];
    if (HAS_LAG) sl[q] = mlag[q];
  }
#endif
  __syncthreads();

  const int lane = threadIdx.x & 31;
  const int wave = threadIdx.x >> 5;
  const int m    = lane & 15;
  const int hb   = lane >> 4;              // 0: lanes 0-15, 1: lanes 16-31
  const int bbase = brow0 + wave * 16;
  const size_t rowstride = (size_t)Tt * Dd;

  // A fragments: 32-bit A 16x4 layout -> vgpr0: K = kb, vgpr1: K = kb+1,
  // where kb = 4*kc + 2*hb  (lanes 16-31 hold K+2).
  v2f a[16], ap[16];
  const float* xr = x + (size_t)(bbase + m) * rowstride + (size_t)t * Dd;
#pragma unroll
  for (int kc = 0; kc < 16; ++kc) a[kc] = *(const v2f*)(xr + kc * 4 + hb * 2);
  if (HAS_LAG) {
    const float* xp = xr - Dd;             // x[b, t-1, :]
#pragma unroll
    for (int kc = 0; kc < 16; ++kc) ap[kc] = *(const v2f*)(xp + kc * 4 + hb * 2);
  }

  v8f acc[4];
#pragma unroll
  for (int nt = 0; nt < 4; ++nt)
#pragma unroll
    for (int v = 0; v < 8; ++v) acc[nt][v] = 0.f;

#pragma unroll
  for (int kc = 0; kc < 16; ++kc) {
    const int kp = kc * 2 + hb;            // pair row index (kb >> 1)
#pragma unroll
    for (int nt = 0; nt < 4; ++nt) {
      const int n = nt * 16 + m;
      // swizzled tile: {B[kb][n], B[kb+1][n]} are adjacent -> one b64 load
      v2f bw = *(const v2f*)(&sw[kp * 128 + n * 2]);
      acc[nt] = __builtin_amdgcn_wmma_f32_16x16x4_f32(
          false, a[kc], false, bw, (short)0, acc[nt], false, false);
      if (HAS_LAG) {
        v2f bl = *(const v2f*)(&sl[kp * 128 + n * 2]);
        acc[nt] = __builtin_amdgcn_wmma_f32_16x16x4_f32(
            false, ap[kc], false, bl, (short)0, acc[nt], false, false);
      }
    }
  }

  // C/D layout: VGPR v holds row M = v + 8*hb, col N = (lane&15) + 16*nt
#pragma unroll
  for (int nt = 0; nt < 4; ++nt)
#pragma unroll
    for (int v = 0; v < 8; ++v) {
      const int mm = v + hb * 8;
      __builtin_nontemporal_store(
          acc[nt][v],
          &out[(size_t)(bbase + mm) * rowstride + (size_t)t * Dd + nt * 16 + m]);
    }
}

// ---------------------------------------------------------------------------
extern "C" void kernel_launch(void* const* d_in, const int* in_sizes, int n_in,
                              void* d_out, int out_size, void* d_ws, size_t ws_size,
                              hipStream_t stream) {
  (void)in_sizes; (void)n_in; (void)out_size; (void)ws_size;
  const float* x    = (const float*)d_in[0];
  const float* iit  = (const float*)d_in[1];
  const float* iis  = (const float*)d_in[2];
  const float* encw = (const float*)d_in[3];
  const float* encb = (const float*)d_in[4];
  const float* l1w  = (const float*)d_in[5];
  const float* l1b  = (const float*)d_in[6];
  const float* l2w  = (const float*)d_in[7];
  const float* l2b  = (const float*)d_in[8];
  const float* d1w  = (const float*)d_in[9];
  const float* d1b  = (const float*)d_in[10];
  const float* d2w  = (const float*)d_in[11];
  const float* d2b  = (const float*)d_in[12];
  const float* d3w  = (const float*)d_in[13];
  const float* d3b  = (const float*)d_in[14];
  const float* uw   = (const float*)d_in[15];
  const float* vw   = (const float*)d_in[16];
  float* out = (float*)d_out;
  float* ws  = (float*)d_ws;

  k_mlag<<<(Dd * Dd + 255) / 256, 256, 0, stream>>>(uw, vw, ws);
  k_odeint<<<1, 1024, 0, stream>>>(iit, iis, encw, encb, l1w, l1b, l2w, l2b,
                                   d1w, d1b, d2w, d2b, d3w, d3b, ws);
  k_tanh<<<(Tt * DK + 255) / 256, 256, 0, stream>>>(ws);
  k_dec1<<<(Tt * Hh + 255) / 256, 256, 0, stream>>>(d1w, d1b, ws);
  k_dec2<<<(Tt * Hh + 255) / 256, 256, 0, stream>>>(d2w, d2b, ws);
  k_dec3<<<(Tt * Dd * Dd + 255) / 256, 256, 0, stream>>>(d3w, d3b, ws);

  dim3 grid0(1, Bb / 128);
  k_apply<false><<<grid0, 256, 0, stream>>>(x, ws, out, 0);   // t = 0 (no lag)
  dim3 grid1(Tt - 1, Bb / 128);
  k_apply<true><<<grid1, 256, 0, stream>>>(x, ws, out, 1);    // t = 1..63
}